// LSTM_26121991094398
// MI455X (gfx1250) — compile-verified
//
#include <hip/hip_runtime.h>
#include <cstdint>
#include <cstddef>

// ---------------- problem constants ----------------
constexpr int IS  = 300;   // input embedding size
constexpr int HS  = 100;   // biLSTM output size
constexpr int H   = 50;    // per-direction hidden
constexpr int G   = 200;   // 4*H gate width
constexpr int Bn  = 256;
constexpr int TN  = 128;
constexpr int NHS = 24;
constexpr int NH  = 6144;  // Bn*NHS
constexpr int TH  = 16;

typedef __attribute__((ext_vector_type(2))) float v2f;
typedef __attribute__((ext_vector_type(8))) float v8f;

#if defined(__gfx1250__) && __has_builtin(__builtin_amdgcn_wmma_f32_16x16x4_f32)
#define USE_WMMA_F32 1
#else
#define USE_WMMA_F32 0
#endif

__device__ __forceinline__ float sigmoidf_(float x) { return 1.f / (1.f + expf(-x)); }

// ---------------------------------------------------------------------------
// C[M,N] = A[M,K] * W[N,K]^T + bias[N]     (M must be a multiple of 128)
// 4 waves / block; block tile 128x16: each wave owns two 16x16 C tiles that
// share the B fragment (8 WMMA per K-chunk). Fast path: one A row per thread,
// 4x global_load_b128 issued before any ds_store_b128 (single wait, no exec
// churn). Rows padded to 20 floats -> 16B-aligned LDS stores, conflict-free.
// ---------------------------------------------------------------------------
__global__ __launch_bounds__(128) void gemm_wmma_bias(
    const float* __restrict__ A, const float* __restrict__ W,
    const float* __restrict__ bias, float* __restrict__ C,
    int M, int N, int K)
{
  __shared__ float sA[128][20];
  __shared__ float sB[16][20];
  const int tid  = threadIdx.x;
  const int wave = tid >> 5;
  const int lane = tid & 31;
  const int m0 = blockIdx.x * 128;
  const int n0 = blockIdx.y * 16;
  const int nl = lane & 15;          // N index (also A's M index per layout)
  const int kb = (lane >> 4) << 1;   // K sub-pair for lanes 16..31
  const int mh = (lane >> 4) << 3;   // C row split at lane 16

  v8f acc0 = {};
  v8f acc1 = {};

  const int brow = tid >> 2, bc4 = tid & 3;     // B tile: threads 0..63
  const bool bactive = (tid < 64) && (n0 + brow < N);

  for (int k0 = 0; k0 < K; k0 += 16) {
    if (k0 + 16 <= K) {
      // ---- fast path: full 16-wide chunk ----
      const float* arow = A + (size_t)(m0 + tid) * K + k0;
      const float4 a0 = *(const float4*)(arow + 0);
      const float4 a1 = *(const float4*)(arow + 4);
      const float4 a2 = *(const float4*)(arow + 8);
      const float4 a3 = *(const float4*)(arow + 12);
      float4 w0 = make_float4(0.f, 0.f, 0.f, 0.f);
      if (bactive)
        w0 = *(const float4*)(W + (size_t)(n0 + brow) * K + k0 + bc4 * 4);
      *(float4*)(&sA[tid][0])  = a0;
      *(float4*)(&sA[tid][4])  = a1;
      *(float4*)(&sA[tid][8])  = a2;
      *(float4*)(&sA[tid][12]) = a3;
      if (tid < 64) *(float4*)(&sB[brow][bc4 * 4]) = w0;
      const int kn = k0 + 16;
      if (kn + 16 <= K)  // prefetch next chunk (global_prefetch_b8)
        __builtin_prefetch(A + (size_t)(m0 + tid) * K + kn, 0, 0);
    } else {
      // ---- tail: zero-padded scalar staging ----
      for (int i = tid; i < 128 * 16; i += 128) {
        const int r = i >> 4, c = i & 15;
        const int kk = k0 + c;
        sA[r][c] = (kk < K) ? A[(size_t)(m0 + r) * K + kk] : 0.f;
      }
      for (int i = tid; i < 16 * 16; i += 128) {
        const int r = i >> 4, c = i & 15;
        const int kk = k0 + c, nn = n0 + r;
        sB[r][c] = (kk < K && nn < N) ? W[(size_t)nn * K + kk] : 0.f;
      }
    }
    __syncthreads();
#if USE_WMMA_F32
    #pragma unroll
    for (int ks = 0; ks < 4; ++ks) {
      v2f a0f, a1f, bf;
      bf.x  = sB[nl][ks * 4 + kb];
      bf.y  = sB[nl][ks * 4 + kb + 1];
      a0f.x = sA[wave * 16 + nl][ks * 4 + kb];
      a0f.y = sA[wave * 16 + nl][ks * 4 + kb + 1];
      a1f.x = sA[64 + wave * 16 + nl][ks * 4 + kb];
      a1f.y = sA[64 + wave * 16 + nl][ks * 4 + kb + 1];
      acc0 = __builtin_amdgcn_wmma_f32_16x16x4_f32(false, a0f, false, bf,
                                                   (short)0, acc0, false, false);
      acc1 = __builtin_amdgcn_wmma_f32_16x16x4_f32(false, a1f, false, bf,
                                                   (short)0, acc1, false, false);
    }
#else
    #pragma unroll
    for (int r = 0; r < 8; ++r) {
      float s0 = acc0[r], s1 = acc1[r];
      for (int c = 0; c < 16; ++c) {
        s0 += sA[wave * 16 + mh + r][c] * sB[nl][c];
        s1 += sA[64 + wave * 16 + mh + r][c] * sB[nl][c];
      }
      acc0[r] = s0; acc1[r] = s1;
    }
#endif
    __syncthreads();
  }

  const int n = n0 + nl;
  if (n < N) {
    const float bv = bias[n];
    #pragma unroll
    for (int r = 0; r < 8; ++r) {
      const int m = m0 + wave * 16 + mh + r;
      C[(size_t)m * N + n] = acc0[r] + bv;
      C[(size_t)(m + 64) * N + n] = acc1[r] + bv;
    }
  }
}

// ---------------------------------------------------------------------------
// Fused LSTM timestep, both directions in one launch (gridDim.y = #dirs).
// Each block: 2 waves own 32 batch rows; g = gx[b,src_t,:] + h_prev*Whh^T via
// 13x13 WMMA k-steps into LDS, then gates + c/h update.
// dir slot 0: forward (lens0 null -> src_t = t)
// dir slot 1: backward (src_t = clip(l[b]-1-t), out = hs_rev, ld = H)
// ---------------------------------------------------------------------------
constexpr int KP  = 52;  // H=50 padded to 13 k-steps of 4
constexpr int KP2 = 54;  // LDS row pitch (8B-aligned rows, conflict-free)

__global__ __launch_bounds__(64) void lstm_step_dual(
    const float* __restrict__ gx0, const float* __restrict__ Whh0,
    float* __restrict__ h0, float* __restrict__ c0,
    float* __restrict__ out0, int ld0, const int* __restrict__ lens0,
    const float* __restrict__ gx1, const float* __restrict__ Whh1,
    float* __restrict__ h1, float* __restrict__ c1,
    float* __restrict__ out1, int ld1, const int* __restrict__ lens1,
    int n, int T, int t)
{
  __shared__ float sH[32][KP2];
  __shared__ float sW[16][KP2];
  __shared__ float sG[32][G];
  __shared__ int   sST[32];

  const float* gx; const float* Whh; float* h_state; float* c_state;
  float* out; int out_ld; const int* lens;
  if (blockIdx.y == 0) {
    gx = gx0; Whh = Whh0; h_state = h0; c_state = c0;
    out = out0; out_ld = ld0; lens = lens0;
  } else {
    gx = gx1; Whh = Whh1; h_state = h1; c_state = c1;
    out = out1; out_ld = ld1; lens = lens1;
  }

  const int tid  = threadIdx.x;
  const int wave = tid >> 5;
  const int lane = tid & 31;
  const int b0 = blockIdx.x * 32;
  const int nl = lane & 15;
  const int kb = (lane >> 4) << 1;
  const int mh = (lane >> 4) << 3;

  // stage h_prev (32 x 52, zero-padded) with float2 loads (rows 8B-aligned)
  for (int i = tid; i < 32 * 26; i += 64) {
    const int r = i / 26, c2 = i % 26;
    float2 v = make_float2(0.f, 0.f);
    if (c2 < 25) v = *(const float2*)(h_state + (size_t)(b0 + r) * H + c2 * 2);
    sH[r][c2 * 2] = v.x; sH[r][c2 * 2 + 1] = v.y;
  }
  if (tid < 32) {
    int st = t;
    if (lens) {
      const int li = lens[b0 + tid];
      st = li - 1 - t;
      st = st < 0 ? 0 : (st > T - 1 ? T - 1 : st);
    }
    sST[tid] = st;
  }

  for (int nt = 0; nt < (G + 15) / 16; ++nt) {  // 13 gate-column tiles
    __syncthreads();
    for (int i = tid; i < 16 * 26; i += 64) {
      const int r = i / 26, c2 = i % 26;
      const int nn = nt * 16 + r;
      float2 v = make_float2(0.f, 0.f);
      if (nn < G && c2 < 25) v = *(const float2*)(Whh + (size_t)nn * H + c2 * 2);
      sW[r][c2 * 2] = v.x; sW[r][c2 * 2 + 1] = v.y;
    }
    __syncthreads();
    v8f acc = {};
#if USE_WMMA_F32
    #pragma unroll
    for (int ks = 0; ks < KP / 4; ++ks) {
      v2f a, b;
      a.x = sH[wave * 16 + nl][ks * 4 + kb];
      a.y = sH[wave * 16 + nl][ks * 4 + kb + 1];
      b.x = sW[nl][ks * 4 + kb];
      b.y = sW[nl][ks * 4 + kb + 1];
      acc = __builtin_amdgcn_wmma_f32_16x16x4_f32(false, a, false, b,
                                                  (short)0, acc, false, false);
    }
#else
    #pragma unroll
    for (int r = 0; r < 8; ++r) {
      float s = 0.f;
      for (int c = 0; c < KP; ++c)
        s += sH[wave * 16 + mh + r][c] * sW[nl][c];
      acc[r] = s;
    }
#endif
    const int ncol = nt * 16 + nl;
    if (ncol < G) {
      #pragma unroll
      for (int r = 0; r < 8; ++r) {
        const int row = wave * 16 + mh + r;
        const int b = b0 + row;
        sG[row][ncol] = acc[r] + gx[((size_t)b * T + sST[row]) * G + ncol];
      }
    }
  }
  __syncthreads();

  for (int idx = tid; idx < 32 * H; idx += 64) {
    const int r = idx / H, j = idx % H;
    const int b = b0 + r;
    const float gi = sG[r][j];
    const float gf = sG[r][H + j];
    const float gg = sG[r][2 * H + j];
    const float go = sG[r][3 * H + j];
    const float cp = c_state[(size_t)b * H + j];
    const float cn = sigmoidf_(gf) * cp + sigmoidf_(gi) * tanhf(gg);
    const float hn = sigmoidf_(go) * tanhf(cn);
    c_state[(size_t)b * H + j] = cn;
    h_state[(size_t)b * H + j] = hn;
    out[((size_t)b * T + t) * out_ld + j] = hn;
  }
}

// Mask forward half by length; un-reverse backward half into cols [H, 2H).
__global__ void finalize_layer(float* __restrict__ out,
                               const float* __restrict__ hs_rev,
                               const int* __restrict__ lens, int n, int T)
{
  size_t idx = (size_t)blockIdx.x * blockDim.x + threadIdx.x;
  size_t total = (size_t)n * T * H;
  if (idx >= total) return;
  int j = (int)(idx % H);
  int t = (int)((idx / H) % T);
  int b = (int)(idx / ((size_t)H * T));
  int l = lens[b];
  float* o = out + ((size_t)b * T + t) * HS;
  if (t < l) {
    int it = l - 1 - t;  // in [0, T-1]
    o[H + j] = hs_rev[((size_t)b * T + it) * H + j];
  } else {
    o[j] = 0.f;
    o[H + j] = 0.f;
  }
}

__global__ void zero_f32(float* __restrict__ p, size_t count)
{
  size_t i = (size_t)blockIdx.x * blockDim.x + threadIdx.x;
  if (i < count) p[i] = 0.f;
}

__global__ void copy_ctx(const float* __restrict__ wn, float* __restrict__ ctx)
{
  int i = blockIdx.x * blockDim.x + threadIdx.x;
  if (i < Bn * HS) {
    int b = i / HS, j = i % HS;
    ctx[i] = wn[(size_t)b * TN * HS + j];
  }
}

// scores[n,t] = h[n,t,:] . (Uw^T q[n]) + q[n].Ub ; masked softmax ; weighted sum
__global__ __launch_bounds__(128) void attention_kernel(
    const float* __restrict__ h,    // (NH, TH, HS)
    const float* __restrict__ q256, // (Bn, HS)
    const float* __restrict__ Uw,   // (HS, HS) row-major [g][hh]
    const float* __restrict__ Ub,   // (HS)
    const int* __restrict__ lens,   // (NH)
    float* __restrict__ out)        // (NH, HS)
{
  __shared__ float sq[HS];
  __shared__ float sr[HS];
  __shared__ float ss[TH];
  __shared__ float sa[TH];
  __shared__ float sred[2];
  const int n = blockIdx.x;
  const int tid = threadIdx.x;
  const int b = n / NHS;
  if (tid < HS) sq[tid] = q256[(size_t)b * HS + tid];
  __syncthreads();
  if (tid < HS) {
    float acc = 0.f;
    for (int g = 0; g < HS; ++g) acc += sq[g] * Uw[(size_t)g * HS + tid];
    sr[tid] = acc;
  }
  if (tid == 0) {
    float c = 0.f;
    for (int g = 0; g < HS; ++g) c += sq[g] * Ub[g];
    sred[0] = c;
  }
  __syncthreads();
  const int L = lens[n];
  if (tid < TH) {
    const float* hp = h + ((size_t)n * TH + tid) * HS;
    float acc = sred[0];
    for (int k = 0; k < HS; ++k) acc += hp[k] * sr[k];
    ss[tid] = (tid < L) ? acc : -1.0e9f;
  }
  __syncthreads();
  if (tid == 0) {
    float mx = ss[0];
    for (int t2 = 1; t2 < TH; ++t2) mx = fmaxf(mx, ss[t2]);
    float sum = 0.f;
    for (int t2 = 0; t2 < TH; ++t2) { float e = expf(ss[t2] - mx); sa[t2] = e; sum += e; }
    sred[1] = 1.f / sum;
  }
  __syncthreads();
  if (tid < HS) {
    const float inv = sred[1];
    float acc = 0.f;
    for (int t2 = 0; t2 < TH; ++t2)
      acc += sa[t2] * h[((size_t)n * TH + t2) * HS + tid];
    out[(size_t)n * HS + tid] = acc * inv;
  }
}

// ---------------------------------------------------------------------------
extern "C" void kernel_launch(void* const* d_in, const int* in_sizes, int n_in,
                              void* d_out, int out_size, void* d_ws, size_t ws_size,
                              hipStream_t stream)
{
  (void)in_sizes; (void)n_in; (void)out_size;
  const float* wemb_n   = (const float*)d_in[0];
  const int*   l_n      = (const int*)d_in[1];
  const float* wemb_hpu = (const float*)d_in[2];
  const int*   l_hpu    = (const int*)d_in[3];
  // d_in[4] = l_hs (unused by the math path)
  const float* encN[12]; const float* encH[12];
  for (int i = 0; i < 12; ++i) {
    encN[i] = (const float*)d_in[5 + i];
    encH[i] = (const float*)d_in[17 + i];
  }
  const float* Uw = (const float*)d_in[29];
  const float* Ub = (const float*)d_in[30];
  const float* Vw = (const float*)d_in[31];
  const float* Vb = (const float*)d_in[32];

  float* wenc_n  = (float*)d_out;                       // (256,128,100)
  float* ctx     = wenc_n + (size_t)Bn * TN * HS;       // (256,100)
  float* wenc_hs = ctx + (size_t)Bn * HS;               // (6144,100) == (256,24,100)

  float* wsf = (float*)d_ws;

  // fused (both directions per launch) needs gx_f and gx_b live together:
  const size_t fused_floats =
      (size_t)2 * NH * TH * HS + (size_t)2 * NH * TH * G +
      (size_t)NH * TH * H + (size_t)4 * NH * H + (size_t)Bn * HS;
  const bool dual = ws_size >= fused_floats * sizeof(float);

  auto run_layer = [&](const float* x, int in_dim, int n, int T, const int* lens,
                       const float* const* p,  // Wih_f, Whh_f, b_f, Wih_b, Whh_b, b_b
                       float* out, float* gxf, float* gxb, float* hsrev, float* st)
  {
    const size_t tokens = (size_t)n * T;
    const size_t nH = (size_t)n * H;
    float* hf = st;           float* cf = st + nH;
    float* hb = st + 2 * nH;  float* cb = st + 3 * nH;
    dim3 gg((unsigned)(tokens / 128), (G + 15) / 16);
    const dim3 sgrid1(n / 32, 1), sgrid2(n / 32, 2);
    if (dual) {
      gemm_wmma_bias<<<gg, 128, 0, stream>>>(x, p[0], p[2], gxf, (int)tokens, G, in_dim);
      gemm_wmma_bias<<<gg, 128, 0, stream>>>(x, p[3], p[5], gxb, (int)tokens, G, in_dim);
      zero_f32<<<(unsigned)((4 * nH + 255) / 256), 256, 0, stream>>>(st, 4 * nH);
      for (int t = 0; t < T; ++t)
        lstm_step_dual<<<sgrid2, 64, 0, stream>>>(
            gxf, p[1], hf, cf, out, HS, nullptr,
            gxb, p[4], hb, cb, hsrev, H, lens, n, T, t);
    } else {
      gemm_wmma_bias<<<gg, 128, 0, stream>>>(x, p[0], p[2], gxf, (int)tokens, G, in_dim);
      zero_f32<<<(unsigned)((2 * nH + 255) / 256), 256, 0, stream>>>(st, 2 * nH);
      for (int t = 0; t < T; ++t)
        lstm_step_dual<<<sgrid1, 64, 0, stream>>>(
            gxf, p[1], hf, cf, out, HS, nullptr,
            gxf, p[1], hf, cf, out, HS, nullptr, n, T, t);
      gemm_wmma_bias<<<gg, 128, 0, stream>>>(x, p[3], p[5], gxf, (int)tokens, G, in_dim);
      zero_f32<<<(unsigned)((2 * nH + 255) / 256), 256, 0, stream>>>(st, 2 * nH);
      for (int t = 0; t < T; ++t)
        lstm_step_dual<<<sgrid1, 64, 0, stream>>>(
            gxf, p[4], hf, cf, hsrev, H, lens,
            gxf, p[4], hf, cf, hsrev, H, lens, n, T, t);
    }
    const size_t tot = tokens * H;
    finalize_layer<<<(unsigned)((tot + 255) / 256), 256, 0, stream>>>(out, hsrev, lens, n, T);
  };

  // -------- encoder over question tokens (n=256, T=128) --------
  {
    size_t o = 0;
    float* L0    = wsf + o; o += (size_t)Bn * TN * HS;
    float* gxf   = wsf + o; o += (size_t)Bn * TN * G;
    float* gxb   = dual ? (wsf + o) : gxf; if (dual) o += (size_t)Bn * TN * G;
    float* hsrev = wsf + o; o += (size_t)Bn * TN * H;
    float* st    = wsf + o; o += (size_t)4 * Bn * H;
    run_layer(wemb_n, IS, Bn, TN, l_n, &encN[0], L0, gxf, gxb, hsrev, st);
    run_layer(L0,    HS, Bn, TN, l_n, &encN[6], wenc_n, gxf, gxb, hsrev, st);
  }
  copy_ctx<<<(Bn * HS + 255) / 256, 256, 0, stream>>>(wenc_n, ctx);

  // -------- encoder over header tokens (n=6144, T=16); reuse arena --------
  float* hC; float* q256;
  {
    size_t o = 0;
    float* hB    = wsf + o; o += (size_t)NH * TH * HS;
    hC           = wsf + o; o += (size_t)NH * TH * HS;
    float* gxf   = wsf + o; o += (size_t)NH * TH * G;
    float* gxb   = dual ? (wsf + o) : gxf; if (dual) o += (size_t)NH * TH * G;
    float* hsrev = wsf + o; o += (size_t)NH * TH * H;
    float* st    = wsf + o; o += (size_t)4 * NH * H;
    q256         = wsf + o; o += (size_t)Bn * HS;
    run_layer(wemb_hpu, IS, NH, TH, l_hpu, &encH[0], hB, gxf, gxb, hsrev, st);
    run_layer(hB,       HS, NH, TH, l_hpu, &encH[6], hC, gxf, gxb, hsrev, st);
  }

  // q = ctx @ Vw^T + Vb   (M=256 = 2 blocks of 128)
  gemm_wmma_bias<<<dim3(Bn / 128, (HS + 15) / 16), 128, 0, stream>>>(ctx, Vw, Vb, q256, Bn, HS, HS);
  attention_kernel<<<NH, 128, 0, stream>>>(hC, q256, Uw, Ub, l_hpu, wenc_hs);
}